// GCN_71451075936964
// MI455X (gfx1250) — compile-verified
//
#include <hip/hip_runtime.h>
#include <hip/hip_bf16.h>

typedef __attribute__((ext_vector_type(2))) float v2f;
typedef __attribute__((ext_vector_type(8))) float v8f;

__device__ __forceinline__ float lrelu(float v) { return v > 0.0f ? v : 0.01f * v; }

// ---------------- degree kernels ----------------
__global__ void gcn_deg_init(float* __restrict__ deg, int n) {
    int i = blockIdx.x * blockDim.x + threadIdx.x;
    if (i < n) deg[i] = 1.0f;   // self-loop
}

__global__ void gcn_deg_count(const int* __restrict__ dst, float* __restrict__ deg, int E) {
    int e = blockIdx.x * blockDim.x + threadIdx.x;
    if (e < E) atomicAdd(&deg[dst[e]], 1.0f);
}

__global__ void gcn_deg_finish(float* __restrict__ dinv, int n) {
    int i = blockIdx.x * blockDim.x + threadIdx.x;
    if (i < n) dinv[i] = rsqrtf(dinv[i]);   // deg >= 1 always (self-loops)
}

// ---------------- WMMA GEMM ----------------
// Computes per 16-row wave tile: T = in[16,K] @ W^T  (W is [NOUT,K] row-major).
// MODE 0: out = lrelu(T + bias)                      (input transform)
// MODE 1: p = T * dinv[row]; out=p; acc_out=p        (conv pre-scale + self-loop acc init)
// Assumes n % 16 == 0 (true: N = 100000 = 6250*16) so tile guards are wave-uniform
// and EXEC is all-ones inside WMMA.
template <int K, int NOUT, int MODE>
__global__ __launch_bounds__(128) void gcn_gemm(
    const float* __restrict__ in,       // [n, K]
    const float* __restrict__ W,        // [NOUT, K]
    const float* __restrict__ bias,     // [NOUT]  (MODE 0)
    const float* __restrict__ dinv,     // [n]     (MODE 1)
    float* __restrict__ out,            // [n, NOUT]
    float* __restrict__ acc_out,        // [n, NOUT] (MODE 1)
    int n)
{
    __shared__ float sW[NOUT * K];
    for (int i = threadIdx.x; i < NOUT * K / 4; i += blockDim.x)
        ((float4*)sW)[i] = ((const float4*)W)[i];
    __syncthreads();

    const int lane  = threadIdx.x & 31;
    const int wave  = threadIdx.x >> 5;
    const int row0  = (blockIdx.x * 4 + wave) * 16;
    if (row0 >= n) return;

    const int lhalf = lane >> 4;    // 0: lanes 0-15, 1: lanes 16-31
    const int l16   = lane & 15;
    const int arow  = row0 + l16;   // A-matrix: lane holds row M = lane%16

    v8f acc[NOUT / 16];
    #pragma unroll
    for (int t = 0; t < NOUT / 16; ++t)
        acc[t] = (v8f){0.f, 0.f, 0.f, 0.f, 0.f, 0.f, 0.f, 0.f};

    #pragma unroll
    for (int k0 = 0; k0 < K; k0 += 4) {
        // A 16x4 f32 layout: v0 = A[M][k0+2*lhalf], v1 = A[M][k0+2*lhalf+1]
        const int kb = k0 + 2 * lhalf;
        v2f a;
        a.x = in[(size_t)arow * K + kb];
        a.y = in[(size_t)arow * K + kb + 1];
        #pragma unroll
        for (int t = 0; t < NOUT / 16; ++t) {
            // B 4x16 f32: B[k][ncol] = W[ncol][k]  (computing in @ W^T)
            const int ncol = t * 16 + l16;
            v2f b;
            b.x = sW[ncol * K + kb];
            b.y = sW[ncol * K + kb + 1];
            acc[t] = __builtin_amdgcn_wmma_f32_16x16x4_f32(
                false, a, false, b, (short)0, acc[t], false, false);
        }
    }

    // C/D layout: VGPR r holds rows r (lanes 0-15) and r+8 (lanes 16-31), col = lane%16
    #pragma unroll
    for (int t = 0; t < NOUT / 16; ++t) {
        const int col = t * 16 + l16;
        if (MODE == 0) {
            const float bv = bias[col];
            #pragma unroll
            for (int r = 0; r < 8; ++r) {
                const int row = row0 + r + 8 * lhalf;
                out[(size_t)row * NOUT + col] = lrelu(acc[t][r] + bv);
            }
        } else {
            #pragma unroll
            for (int r = 0; r < 8; ++r) {
                const int row = row0 + r + 8 * lhalf;
                const float val = acc[t][r] * dinv[row];
                out[(size_t)row * NOUT + col]     = val;   // p (gather source)
                acc_out[(size_t)row * NOUT + col] = val;   // acc init = self-loop term
            }
        }
    }
}

// ---------------- edge scatter: acc[dst] += p[src], 32 channels ----------------
// 8 threads per edge, 4 channels each (float4 gather + 4 f32 atomics into L2-resident acc)
__global__ void gcn_scatter(const int* __restrict__ src, const int* __restrict__ dst,
                            const float* __restrict__ p, float* __restrict__ acc, int E)
{
    int t = blockIdx.x * blockDim.x + threadIdx.x;
    int e = t >> 3;
    if (e >= E) return;
    int c4 = (t & 7) * 4;
    int u = src[e];
    int v = dst[e];
    float4 m = *(const float4*)(p + (size_t)u * 32 + c4);
    float* a = acc + (size_t)v * 32 + c4;
    atomicAdd(a + 0, m.x);
    atomicAdd(a + 1, m.y);
    atomicAdd(a + 2, m.z);
    atomicAdd(a + 3, m.w);
}

// ---------------- h = lrelu(dinv[v]*acc + b), in place ----------------
__global__ void gcn_finish(float* __restrict__ acc, const float* __restrict__ dinv,
                           const float* __restrict__ bias, int n)
{
    int t = blockIdx.x * blockDim.x + threadIdx.x;
    if (t >= n * 32) return;
    int row = t >> 5;
    int c   = t & 31;
    acc[t] = lrelu(dinv[row] * acc[t] + bias[c]);
}

// ---------------- final: out[v] = W_out . lrelu(dinv*acc2 + b2) + b_out ----------------
__global__ void gcn_out(const float* __restrict__ acc2, const float* __restrict__ dinv,
                        const float* __restrict__ b2, const float* __restrict__ Wout,
                        const float* __restrict__ bout, float* __restrict__ out, int n)
{
    int i = blockIdx.x * blockDim.x + threadIdx.x;
    if (i >= n) return;
    float di = dinv[i];
    float s = 0.0f;
    #pragma unroll
    for (int c = 0; c < 32; ++c)
        s += Wout[c] * lrelu(di * acc2[(size_t)i * 32 + c] + b2[c]);
    out[i] = s + bout[0];
}

extern "C" void kernel_launch(void* const* d_in, const int* in_sizes, int n_in,
                              void* d_out, int out_size, void* d_ws, size_t ws_size,
                              hipStream_t stream)
{
    const float* x     = (const float*)d_in[0];
    const int*   ei    = (const int*)d_in[1];   // edge_index [2,E]
    const float* W_in  = (const float*)d_in[2];
    const float* b_in  = (const float*)d_in[3];
    const float* W1    = (const float*)d_in[4];
    const float* b1    = (const float*)d_in[5];
    const float* W2    = (const float*)d_in[6];
    const float* b2    = (const float*)d_in[7];
    const float* W_out = (const float*)d_in[8];
    const float* b_out = (const float*)d_in[9];
    float* out = (float*)d_out;

    const int N = in_sizes[0] / 64;
    const int E = in_sizes[1] / 2;
    const int* src = ei;
    const int* dst = ei + E;

    // workspace layout (floats): dinv[N] | h0[64N] | p[32N] | accA[32N] | accB[32N]
    float* ws   = (float*)d_ws;
    float* dinv = ws;
    float* h0   = ws + (size_t)N;
    float* p    = ws + (size_t)N * 65;
    float* accA = ws + (size_t)N * 97;
    float* accB = ws + (size_t)N * 129;

    const int TB = 256;
    // 1) degrees -> dinv
    gcn_deg_init  <<<(N + TB - 1) / TB, TB, 0, stream>>>(dinv, N);
    gcn_deg_count <<<(E + TB - 1) / TB, TB, 0, stream>>>(dst, dinv, E);
    gcn_deg_finish<<<(N + TB - 1) / TB, TB, 0, stream>>>(dinv, N);

    // GEMM grid: 16-row tiles, 4 waves (tiles) per 128-thread block
    const int tiles  = (N + 15) / 16;
    const int gblk   = (tiles + 3) / 4;

    // 2) h0 = lrelu(x @ W_in^T + b_in)
    gcn_gemm<64, 64, 0><<<gblk, 128, 0, stream>>>(x, W_in, b_in, nullptr, h0, nullptr, N);

    // 3) conv1: p = (h0 @ W1^T)*dinv ; accA = p(self-loop) ; scatter ; finish
    gcn_gemm<64, 32, 1><<<gblk, 128, 0, stream>>>(h0, W1, nullptr, dinv, p, accA, N);
    gcn_scatter<<<((size_t)E * 8 + TB - 1) / TB, TB, 0, stream>>>(src, dst, p, accA, E);
    gcn_finish <<<((size_t)N * 32 + TB - 1) / TB, TB, 0, stream>>>(accA, dinv, b1, N);

    // 4) conv2: p = (h1 @ W2^T)*dinv ; accB = p ; scatter
    gcn_gemm<32, 32, 1><<<gblk, 128, 0, stream>>>(accA, W2, nullptr, dinv, p, accB, N);
    gcn_scatter<<<((size_t)E * 8 + TB - 1) / TB, TB, 0, stream>>>(src, dst, p, accB, E);

    // 5) out = W_out . lrelu(dinv*accB + b2) + b_out
    gcn_out<<<(N + TB - 1) / TB, TB, 0, stream>>>(accB, dinv, b2, W_out, b_out, out, N);
}